// ModelWrapper_21680994910706
// MI455X (gfx1250) — compile-verified
//
#include <hip/hip_runtime.h>
#include <hip/hip_bf16.h>
#include <stdint.h>

// ---------------- problem constants (from reference) ----------------
#define N_BOX   25200
#define N_CLS   80
#define PSTRIDE 85          // 4 box + 1 obj + 80 cls
#define TOPK    5000
#define SORT_M  32768       // next pow2 >= N_BOX
#define CONF_T  0.25f
#define IOU_T   0.45f
#define SCORE_TPB 256       // boxes (threads) per score block

// ---------------- CDNA5 TDM availability probes ----------------
#if defined(__has_builtin)
#  if __has_builtin(__builtin_amdgcn_tensor_load_to_lds) && __has_builtin(__builtin_amdgcn_s_wait_tensorcnt)
#    define USE_TDM 1
#  endif
#  if __has_builtin(__builtin_amdgcn_tensor_store_from_lds) && __has_builtin(__builtin_amdgcn_s_wait_tensorcnt)
#    define USE_TDM_ST 1
#  endif
#endif

#ifdef USE_TDM
#pragma message("gfx1250: TDM tensor_load_to_lds path ENABLED")
#else
#pragma message("gfx1250: TDM load builtin not found; using VALU copy fallback")
#endif
#ifdef USE_TDM_ST
#pragma message("gfx1250: TDM tensor_store_from_lds path ENABLED")
#else
#pragma message("gfx1250: TDM store builtin not found; using VALU store fallback")
#endif

typedef unsigned int v4u __attribute__((ext_vector_type(4)));
typedef int          v4i __attribute__((ext_vector_type(4)));
typedef int          v8i __attribute__((ext_vector_type(8)));

// --------------------------------------------------------------------
// Shared 1-row D# builder: `ndwords` contiguous dwords, LDS <-> global.
// group0: count=1 | lds_addr | global_addr[56:0] | type=2
// group1: data_size=4B, tensor_dim0 = tile_dim0 = ndwords (<=65535),
//         tensor_dim1 = tile_dim1 = 1, dim0_stride = ndwords.
// Caller: issue from one wave, then s_wait_tensorcnt(0) + __syncthreads().
// --------------------------------------------------------------------
#if defined(USE_TDM) || defined(USE_TDM_ST)
__device__ __forceinline__ void tdm_build_1d(v4u& g0, v8i& g1,
                                             const void* lds_ptr,
                                             const void* gptr, int ndwords) {
    unsigned lds_addr = (unsigned)(unsigned long long)(uintptr_t)lds_ptr;
    unsigned long long ga = (unsigned long long)(uintptr_t)gptr;
    g0[0] = 1u;
    g0[1] = lds_addr;
    g0[2] = (unsigned)(ga & 0xFFFFFFFFull);
    g0[3] = (unsigned)((ga >> 32) & 0x01FFFFFFull) | (2u << 30);
    g1[0] = (int)(2u << 16);                                // data_size = 4B
    g1[1] = (int)((ndwords & 0xFFFF) << 16);                // tensor_dim0 lo16
    g1[2] = (int)(((ndwords >> 16) & 0xFFFF) | (1 << 16));  // dim0 hi | dim1=1
    g1[3] = (int)((ndwords & 0xFFFF) << 16);                // tile_dim0
    g1[4] = 1;                                              // tile_dim1 = 1
    g1[5] = ndwords;                                        // dim0_stride
    g1[6] = 0;
    g1[7] = 0;
}
#endif

#ifdef USE_TDM
__device__ __forceinline__ void tdm_load_1d(void* lds_dst, const void* gsrc,
                                            int ndwords) {
    v4u g0; v8i g1;
    tdm_build_1d(g0, g1, lds_dst, gsrc, ndwords);
    v4i gz = {0, 0, 0, 0};
#if defined(__clang_major__) && (__clang_major__ >= 23)
    v8i gz8 = {0, 0, 0, 0, 0, 0, 0, 0};
    __builtin_amdgcn_tensor_load_to_lds(g0, g1, gz, gz, gz8, 0);
#else
    __builtin_amdgcn_tensor_load_to_lds(g0, g1, gz, gz, 0);
#endif
}
#endif

#ifdef USE_TDM_ST
__device__ __forceinline__ void tdm_store_1d(const void* lds_src, void* gdst,
                                             int ndwords) {
    v4u g0; v8i g1;
    tdm_build_1d(g0, g1, lds_src, gdst, ndwords);
    v4i gz = {0, 0, 0, 0};
#if defined(__clang_major__) && (__clang_major__ >= 23)
    v8i gz8 = {0, 0, 0, 0, 0, 0, 0, 0};
    __builtin_amdgcn_tensor_store_from_lds(g0, g1, gz, gz, gz8, 0);
#else
    __builtin_amdgcn_tensor_store_from_lds(g0, g1, gz, gz, 0);
#endif
}
#endif

// ====================================================================
// Stage 1: per-box score = obj * max(cls), argmax class, xywh->xyxy.
// HBM-bound (~68.6 MB -> ~3us @ 23.3 TB/s). Each block TDM-stages its
// 256x85-float tile (87 KB LDS) as ONE contiguous DMA, then reduces
// from LDS (stride 85 odd -> coprime with 64 banks -> conflict free).
// ====================================================================
__global__ void nms_score_kernel(const float* __restrict__ pred,
                                 float* __restrict__ mscore,
                                 float* __restrict__ clsv,
                                 float* __restrict__ boxes,
                                 int total) {
    extern __shared__ float tile[];                 // SCORE_TPB * 85 floats
    const int base = blockIdx.x * SCORE_TPB;
    const int cnt  = (total - base < SCORE_TPB) ? (total - base) : SCORE_TPB;
    const int nd   = cnt * PSTRIDE;

    // prefetch a slice of the next block's tile into L2 (global_prefetch_b8)
    if (base + SCORE_TPB < total)
        __builtin_prefetch(pred + (size_t)(base + SCORE_TPB) * PSTRIDE +
                               (size_t)threadIdx.x * 16, 0, 1);

#ifdef USE_TDM
    if (threadIdx.x < 32) {
        tdm_load_1d(tile, pred + (size_t)base * PSTRIDE, nd);
        __builtin_amdgcn_s_wait_tensorcnt(0);
    }
#else
    for (int t = threadIdx.x; t < nd; t += SCORE_TPB)
        tile[t] = pred[(size_t)base * PSTRIDE + t];
#endif
    __syncthreads();

    if ((int)threadIdx.x >= cnt) return;
    const float* p = tile + threadIdx.x * PSTRIDE;
    float x = p[0], y = p[1], w = p[2], h = p[3], obj = p[4];
    float best = -1.0f;
    int bi = 0;
#pragma unroll 8
    for (int c = 0; c < N_CLS; ++c) {
        float v = p[5 + c] * obj;
        if (v > best) { best = v; bi = c; }   // first-max, matches jnp.argmax
    }
    const int g = base + threadIdx.x;
    float hw = 0.5f * w, hh = 0.5f * h;
    boxes[(size_t)g * 4 + 0] = x - hw;
    boxes[(size_t)g * 4 + 1] = y - hh;
    boxes[(size_t)g * 4 + 2] = x + hw;
    boxes[(size_t)g * 4 + 3] = y + hh;
    clsv[g]   = (float)bi;
    mscore[g] = ((obj > CONF_T) && (best > CONF_T)) ? best : -1.0f;
}

// ====================================================================
// Stage 2: per-batch descending bitonic sort of 32768 (score,idx) pairs
// entirely in LDS (256 KB of CDNA5's 320 KB WGP LDS). One 32-wave
// workgroup per batch; scores arrive via one 25200-dword TDM DMA while
// the waves initialize padding + index array (disjoint LDS regions).
// ====================================================================
__global__ void nms_sort_kernel(const float* __restrict__ mscore,
                                const float* __restrict__ boxes,
                                const float* __restrict__ clsv,
                                float* __restrict__ sdata) {
    extern __shared__ unsigned char smem[];
    float* skey = (float*)smem;                       // 32768 * 4B
    int*   sval = (int*)(smem + (size_t)SORT_M * 4);  // 32768 * 4B
    const int b    = blockIdx.x;
    const int tid  = threadIdx.x;
    const int nthr = blockDim.x;

#ifdef USE_TDM
    if (tid < 32) {
        tdm_load_1d(skey, mscore + (size_t)b * N_BOX, N_BOX);
        __builtin_amdgcn_s_wait_tensorcnt(0);
    }
    for (int i = tid; i < SORT_M; i += nthr) {
        if (i >= N_BOX) skey[i] = -3.0e38f;
        sval[i] = i;
    }
#else
    for (int i = tid; i < SORT_M; i += nthr) {
        skey[i] = (i < N_BOX) ? mscore[(size_t)b * N_BOX + i] : -3.0e38f;
        sval[i] = i;
    }
#endif
    __syncthreads();

    // descending bitonic sort
    for (int k = 2; k <= SORT_M; k <<= 1) {
        for (int j = k >> 1; j > 0; j >>= 1) {
            for (int t = tid; t < SORT_M / 2; t += nthr) {
                int i = ((t & ~(j - 1)) << 1) | (t & (j - 1));
                int p = i | j;
                bool desc = ((i & k) == 0);
                float a = skey[i], c2 = skey[p];
                if (desc ? (a < c2) : (a > c2)) {
                    skey[i] = c2; skey[p] = a;
                    int tv = sval[i]; sval[i] = sval[p]; sval[p] = tv;
                }
            }
            __syncthreads();
        }
    }

    for (int t = tid; t < TOPK; t += nthr) {
        int idx  = sval[t];
        float s  = skey[t];
        size_t src = ((size_t)b * N_BOX + idx) * 4;
        float* dst = sdata + ((size_t)b * TOPK + t) * 6;
        dst[0] = boxes[src + 0];
        dst[1] = boxes[src + 1];
        dst[2] = boxes[src + 2];
        dst[3] = boxes[src + 3];
        dst[4] = s;
        dst[5] = clsv[(size_t)b * N_BOX + idx];
    }
}

// ====================================================================
// Stage 3: greedy IoU suppression. One 32-wave workgroup per batch.
// 5000x6 tile (120 KB) + keep flags (20 KB) resident in LDS, staged in
// by one 30000-dword TDM DMA; keep flags written back by one TDM store.
// Class offset (+cls*7680) reduces exactly to class equality since all
// coords < 7680. Latency-limited by the inherent 5000-step sequential
// dependency of greedy NMS.
// ====================================================================
__global__ void nms_suppress_kernel(const float* __restrict__ sdata,
                                    unsigned* __restrict__ keepg) {
    extern __shared__ unsigned char smem2[];
    float*    sb   = (float*)smem2;                               // TOPK*6
    unsigned* keep = (unsigned*)(smem2 + (size_t)TOPK * 6 * 4);   // TOPK
    const int b    = blockIdx.x;
    const int tid  = threadIdx.x;
    const int nthr = blockDim.x;

#ifdef USE_TDM
    if (tid < 32) {
        tdm_load_1d(sb, sdata + (size_t)b * TOPK * 6, TOPK * 6);
        __builtin_amdgcn_s_wait_tensorcnt(0);
    }
#else
    for (int t = tid; t < TOPK * 6; t += nthr)
        sb[t] = sdata[(size_t)b * TOPK * 6 + t];
#endif
    __syncthreads();

    for (int t = tid; t < TOPK; t += nthr)
        keep[t] = (sb[t * 6 + 4] > 0.0f) ? 1u : 0u;
    __syncthreads();

    for (int i = 0; i < TOPK - 1; ++i) {
        if (keep[i]) {   // uniform across block: no writes to keep[i] this step
            const float x1 = sb[i * 6 + 0], y1 = sb[i * 6 + 1];
            const float x2 = sb[i * 6 + 2], y2 = sb[i * 6 + 3];
            const float ci = sb[i * 6 + 5];
            const float ai = (x2 - x1) * (y2 - y1);
            for (int j = i + 1 + tid; j < TOPK; j += nthr) {
                if (keep[j] && sb[j * 6 + 5] == ci) {
                    float u1 = sb[j * 6 + 0], v1 = sb[j * 6 + 1];
                    float u2 = sb[j * 6 + 2], v2 = sb[j * 6 + 3];
                    float aj = (u2 - u1) * (v2 - v1);
                    float iw = fminf(x2, u2) - fmaxf(x1, u1);
                    float ih = fminf(y2, v2) - fmaxf(y1, v1);
                    iw = fmaxf(iw, 0.0f);
                    ih = fmaxf(ih, 0.0f);
                    float inter = iw * ih;
                    float iou = inter / (ai + aj - inter + 1e-7f);
                    if (iou > IOU_T) keep[j] = 0u;
                }
            }
            __syncthreads();
        }
    }

    __syncthreads();   // all keep[] writes LDS-visible before DMA-out
#ifdef USE_TDM_ST
    if (tid < 32) {
        tdm_store_1d(keep, keepg + (size_t)b * TOPK, TOPK);
        __builtin_amdgcn_s_wait_tensorcnt(0);
    }
#else
    for (int t = tid; t < TOPK; t += nthr)
        keepg[(size_t)b * TOPK + t] = keep[t];
#endif
}

// ====================================================================
// Stage 4: compact kept boxes (already descending by score, so
// compaction == final top_k) into out[b][max_det][6]; zero the rest.
// ====================================================================
__global__ void nms_output_kernel(const float* __restrict__ sdata,
                                  const unsigned* __restrict__ keepg,
                                  const int* __restrict__ max_det_p,
                                  float* __restrict__ out,
                                  int rows) {
    const int b   = blockIdx.x;
    const int tid = threadIdx.x;
    float* obase = out + (size_t)b * rows * 6;
    for (int t = tid; t < rows * 6; t += blockDim.x) obase[t] = 0.0f;
    __syncthreads();
    if (tid == 0) {
        int md = *max_det_p;
        if (md > rows) md = rows;
        int cnt = 0;
        for (int k = 0; k < TOPK && cnt < md; ++k) {
            if (keepg[(size_t)b * TOPK + k]) {
                const float* src = sdata + ((size_t)b * TOPK + k) * 6;
                float* dst = obase + (size_t)cnt * 6;
                dst[0] = src[0]; dst[1] = src[1]; dst[2] = src[2];
                dst[3] = src[3]; dst[4] = src[4]; dst[5] = src[5];
                ++cnt;
            }
        }
    }
}

// ====================================================================
extern "C" void kernel_launch(void* const* d_in, const int* in_sizes, int n_in,
                              void* d_out, int out_size, void* d_ws, size_t ws_size,
                              hipStream_t stream) {
    (void)n_in; (void)ws_size;
    const float* pred   = (const float*)d_in[0];
    const int*   maxdet = (const int*)d_in[1];
    float*       out    = (float*)d_out;

    const int B    = in_sizes[0] / (N_BOX * PSTRIDE);
    const int rows = (B > 0) ? (out_size / (B * 6)) : 0;

    // workspace carve (all 256B aligned): ~6 MB total
    uint8_t* ws = (uint8_t*)d_ws;
    size_t off = 0;
    auto carve = [&](size_t bytes) {
        uint8_t* p = ws + off;
        off += (bytes + 255) & ~(size_t)255;
        return p;
    };
    float*    mscore = (float*)carve((size_t)B * N_BOX * 4);
    float*    clsv   = (float*)carve((size_t)B * N_BOX * 4);
    float*    boxes  = (float*)carve((size_t)B * N_BOX * 16);
    float*    sdata  = (float*)carve((size_t)B * TOPK * 6 * 4);
    unsigned* keepg  = (unsigned*)carve((size_t)B * TOPK * 4);

    const int total = B * N_BOX;

    const size_t scoreLds = (size_t)SCORE_TPB * PSTRIDE * 4;   // 87,040 B
    hipFuncSetAttribute((const void*)nms_score_kernel,
                        hipFuncAttributeMaxDynamicSharedMemorySize, (int)scoreLds);
    nms_score_kernel<<<(total + SCORE_TPB - 1) / SCORE_TPB, SCORE_TPB,
                       scoreLds, stream>>>(pred, mscore, clsv, boxes, total);

    const size_t sortLds = (size_t)SORT_M * 8;                 // 256 KB
    hipFuncSetAttribute((const void*)nms_sort_kernel,
                        hipFuncAttributeMaxDynamicSharedMemorySize, (int)sortLds);
    nms_sort_kernel<<<B, 1024, sortLds, stream>>>(mscore, boxes, clsv, sdata);

    const size_t nmsLds = (size_t)TOPK * 6 * 4 + (size_t)TOPK * 4;  // 140 KB
    hipFuncSetAttribute((const void*)nms_suppress_kernel,
                        hipFuncAttributeMaxDynamicSharedMemorySize, (int)nmsLds);
    nms_suppress_kernel<<<B, 1024, nmsLds, stream>>>(sdata, keepg);

    nms_output_kernel<<<B, 256, 0, stream>>>(sdata, keepg, maxdet, out, rows);
}